// Matcher_55430847922718
// MI455X (gfx1250) — compile-verified
//
#include <hip/hip_runtime.h>

// Problem constants (from the reference): B=32, Q=1024, G=64 (C=256 unused).
#define NB 32
#define NQ 1024
#define NG 64
#define ROWPAD 65          // LDS row pitch in floats: (j*65+i)%64 == (j+i)%64 -> conflict-free column scans
#define NT 256             // solver threads per block (8 wave32 waves)
#define COLS_PER_T (NQ / NT)

#define COST_OBJ 5.0f
#define COST_CTR 10.0f
#define COST_GIOU 2.0f

// ---- gfx1250 CDNA5 data-movement feature detection ----
#if defined(__has_builtin)
#if __has_builtin(__builtin_amdgcn_tensor_load_to_lds)
#define HAVE_TDM 1
#endif
#if __has_builtin(__builtin_amdgcn_global_load_async_to_lds_b128)
#define HAVE_ASYNC_B128 1
#endif
#if __has_builtin(__builtin_amdgcn_s_wait_asynccnt)
#define HAVE_WAIT_ASYNC 1
#endif
#endif

// ROCm 7.2 builtin is 5-arg; amdgpu-toolchain (which uniquely ships the TDM
// header) is 6-arg. Key the arity off the header's presence.
#if defined(__has_include)
#if __has_include(<hip/amd_detail/amd_gfx1250_TDM.h>)
#define TDM_SIX_ARGS 1
#endif
#endif

typedef unsigned int u32x4 __attribute__((vector_size(16)));
typedef int i32x4 __attribute__((vector_size(16)));
typedef int i32x8 __attribute__((vector_size(32)));

// async b128 fallback param types (pointee = 16B int vector, global/shared AS)
typedef int v4i_ __attribute__((vector_size(16)));
typedef __attribute__((address_space(1))) v4i_ GlobV4;
typedef __attribute__((address_space(3))) v4i_ LdsV4;
typedef __attribute__((address_space(3))) void LdsVoidT;

// Kernel 1: build cost[b][q][i] = -5*obj[b,q] + 10*cd[b,q,i] - 2*giou[b,q,i]
// Flattened index e = ((b*NQ)+q)*NG + i  ->  fully coalesced reads and writes.
__global__ void build_cost_kernel(const float* __restrict__ obj,
                                  const float* __restrict__ cd,
                                  const float* __restrict__ gi,
                                  float* __restrict__ costws) {
  int e = blockIdx.x * blockDim.x + threadIdx.x;
  if (e < NB * NQ * NG) {
    float o = obj[e >> 6];  // e>>6 == b*NQ + q
    costws[e] = -COST_OBJ * o + COST_CTR * cd[e] - COST_GIOU * gi[e];
  }
}

// Kernel 2: one workgroup per batch. Whole cost tile + JV state resident in LDS.
// Exact Jonker-Volgenant on the transposed (g x 1024) problem, matching the
// reference's transposed branch: row4col[q] = matched GT row (or -1).
template <bool USE_WS>
__global__ __launch_bounds__(NT) void hungarian_kernel(
    const float* __restrict__ obj, const float* __restrict__ cd,
    const float* __restrict__ gi, const int* __restrict__ ngt,
    const float* __restrict__ costws, float* __restrict__ out) {
  const int b = blockIdx.x;
  const int t = threadIdx.x;

  extern __shared__ float smem[];
  float* costL    = smem;                    // [NQ][ROWPAD]  256KB+pad
  float* shortest = costL + NQ * ROWPAD;     // [NQ]
  float* vdual    = shortest + NQ;           // [NQ]
  int*   path     = (int*)(vdual + NQ);      // [NQ]
  int*   row4col  = path + NQ;               // [NQ]
  int*   scset    = row4col + NQ;            // [NQ]

  __shared__ float u_s[NG];
  __shared__ int   col4row[NG];
  __shared__ int   srflag[NG];
  __shared__ float redv[NT / 32];
  __shared__ int   redi[NT / 32];
  __shared__ float sh_minval;
  __shared__ int   sh_i;
  __shared__ int   sh_sink;

  int g = ngt[b];
  if (g < 0) g = 0;
  if (g > NG) g = NG;

  // ---- Stage the 1024x64 cost tile into LDS (padded rows, pitch = 65 floats) ----
  if (USE_WS) {
    const float* src = costws + (size_t)b * NQ * NG;
#if defined(HAVE_TDM)
    // Tensor Data Mover: 2 descriptors (waves 0 and 1), each DMAs a 512x64 f32
    // sub-tile. Hardware padding inserts 1 DWORD after every 64 DWORDs
    // (pad_interval=5 -> 64 DW, pad_amount=0 -> 1 DW), producing the 65-float
    // LDS pitch directly. Tensor ops are per-wave and EXEC-ignored; the wv<2
    // guard is wave-uniform so waves 2..7 branch around the issue.
    const int wv = t >> 5;
    if (wv < 2) {
      unsigned long long ga =
          (unsigned long long)(uintptr_t)(src + (size_t)wv * (NQ / 2) * NG);
      unsigned lb = (unsigned)(uintptr_t)(LdsVoidT*)(costL + wv * (NQ / 2) * ROWPAD);
      // D# group 0: count=1 | lds_addr | global_addr[56:0] | type=2
      u32x4 g0 = {1u, lb, (unsigned)ga,
                  ((unsigned)((ga >> 32) & 0x01FFFFFFull)) | 0x80000000u};
      // D# group 1: data_size=4B, pad_enable, pad_interval=64DW, pad_amount=1DW,
      // tensor_dim0=64, tensor_dim1=512, tile_dim0=64, tile_dim1=512,
      // tensor_dim0_stride=64. workgroup_mask=0 (not in a cluster).
      i32x8 g1 = {(int)((2u << 16) | (1u << 20) | (5u << 22)),  // 0x01520000
                  (int)(64u << 16),
                  (int)((unsigned)(NQ / 2) << 16),
                  (int)(64u << 16),
                  (int)(NQ / 2),
                  64, 0, 0};
      i32x4 gz = {0, 0, 0, 0};
#if defined(TDM_SIX_ARGS)
      i32x8 gz8 = {0, 0, 0, 0, 0, 0, 0, 0};
      __builtin_amdgcn_tensor_load_to_lds(g0, g1, gz, gz, gz8, 0);
#else
      __builtin_amdgcn_tensor_load_to_lds(g0, g1, gz, gz, 0);
#endif
      __builtin_amdgcn_s_wait_tensorcnt(0);
    }
#elif defined(HAVE_ASYNC_B128)
    // Async global->LDS copy: 16384 chunks of 16B, 16 chunks per 64-float row;
    // consecutive threads touch consecutive 16B chunks (coalesced), LDS dest
    // keeps within-row contiguity (pad only between rows).
    for (int c = t; c < (NQ * NG) / 4; c += NT) {
      int q = c >> 4;
      int k = (c & 15) << 2;
      __builtin_amdgcn_global_load_async_to_lds_b128(
          (GlobV4*)(src + q * NG + k),
          (LdsV4*)(costL + q * ROWPAD + k), 0, 0);
    }
#if defined(HAVE_WAIT_ASYNC)
    __builtin_amdgcn_s_wait_asynccnt(0);
#else
    asm volatile("s_wait_asynccnt 0" ::: "memory");
#endif
#else
    for (int c = t; c < (NQ * NG) / 4; c += NT) {
      int q = c >> 4;
      int k = (c & 15) << 2;
      const float4 v = *(const float4*)(src + q * NG + k);
      *(float4*)(costL + q * ROWPAD + k) = v;
    }
#endif
  } else {
    // Fallback: compute the tile directly into LDS (each thread owns whole rows).
    for (int q = t; q < NQ; q += NT) {
      float base = -COST_OBJ * obj[b * NQ + q];
      const float* cdr = cd + ((size_t)b * NQ + q) * NG;
      const float* gir = gi + ((size_t)b * NQ + q) * NG;
      for (int i = 0; i < NG; ++i)
        costL[q * ROWPAD + i] = base + COST_CTR * cdr[i] - COST_GIOU * gir[i];
    }
  }

  // ---- Solver state init ----
  for (int jj = 0; jj < COLS_PER_T; ++jj) {
    int j = t + jj * NT;
    row4col[j] = -1;
    vdual[j] = 0.0f;
  }
  if (t < NG) { u_s[t] = 0.0f; col4row[t] = -1; }
  __syncthreads();

  const float INF = __builtin_inff();

  for (int cur = 0; cur < g; ++cur) {
    // Phase A: per-augmentation reset
    for (int jj = 0; jj < COLS_PER_T; ++jj) {
      int j = t + jj * NT;
      shortest[j] = INF;
      path[j] = -1;
      scset[j] = 0;
    }
    if (t < NG) srflag[t] = 0;
    if (t == 0) { sh_minval = 0.0f; sh_i = cur; sh_sink = -1; }
    __syncthreads();
    if (t == 0) srflag[cur] = 1;  // only read in phase C (post-barrier)

    // Phase B: Dijkstra shortest augmenting path (uniform loop)
    for (;;) {
      int i = sh_i;
      float minv = sh_minval;
      float ui = u_s[i];
      float bv = INF;
      unsigned bj = 0xFFFFFFFFu;
      for (int jj = 0; jj < COLS_PER_T; ++jj) {
        int j = t + jj * NT;
        if (!scset[j]) {
          float r = minv + costL[j * ROWPAD + i] - ui - vdual[j];
          float s = shortest[j];
          if (r < s) { shortest[j] = r; path[j] = i; s = r; }
          if (s < bv || (s == bv && (unsigned)j < bj)) { bv = s; bj = (unsigned)j; }
        }
      }
      // wave32 argmin (value, then smallest column index = argmin-first semantics)
      for (int off = 16; off > 0; off >>= 1) {
        float ov = __shfl_xor(bv, off, 32);
        unsigned oj = (unsigned)__shfl_xor((int)bj, off, 32);
        if (ov < bv || (ov == bv && oj < bj)) { bv = ov; bj = oj; }
      }
      if ((t & 31) == 0) { redv[t >> 5] = bv; redi[t >> 5] = (int)bj; }
      __syncthreads();
      if (t == 0) {
        float mv = redv[0];
        unsigned mj = (unsigned)redi[0];
        for (int w = 1; w < NT / 32; ++w) {
          float ov = redv[w];
          unsigned oj = (unsigned)redi[w];
          if (ov < mv || (ov == mv && oj < mj)) { mv = ov; mj = oj; }
        }
        scset[mj] = 1;
        sh_minval = mv;
        int r4 = row4col[mj];
        if (r4 < 0) sh_sink = (int)mj;
        else { sh_i = r4; srflag[r4] = 1; }
      }
      __syncthreads();
      if (sh_sink >= 0) break;
    }

    // Phase C: dual updates (read pre-augmentation col4row), then augment
    float minval = sh_minval;
    if (t < g && srflag[t]) {
      if (t == cur) u_s[t] += minval;
      else u_s[t] += minval - shortest[col4row[t]];
    }
    for (int jj = 0; jj < COLS_PER_T; ++jj) {
      int j = t + jj * NT;
      if (scset[j]) vdual[j] -= minval - shortest[j];
    }
    __syncthreads();
    if (t == 0) {
      int j = sh_sink;
      for (;;) {
        int i = path[j];
        row4col[j] = i;
        int jn = col4row[i];
        col4row[i] = j;
        j = jn;
        if (i == cur) break;
      }
    }
    __syncthreads();
  }

  // Outputs: d_out = [per_prop_gt_inds (B*Q floats) | matched_mask (B*Q floats)]
  float* outInds = out;
  float* outMask = out + NB * NQ;
  for (int jj = 0; jj < COLS_PER_T; ++jj) {
    int j = t + jj * NT;
    int r = row4col[j];
    outInds[b * NQ + j] = (r >= 0) ? (float)r : 0.0f;
    outMask[b * NQ + j] = (r >= 0) ? 1.0f : 0.0f;
  }
}

extern "C" void kernel_launch(void* const* d_in, const int* in_sizes, int n_in,
                              void* d_out, int out_size, void* d_ws, size_t ws_size,
                              hipStream_t stream) {
  (void)in_sizes; (void)n_in; (void)out_size;
  // d_in[0] = sem_cls_prob (unused: class_mat == 0 in the reference)
  const float* obj = (const float*)d_in[1];
  const float* cd  = (const float*)d_in[2];
  const float* gi  = (const float*)d_in[3];
  const int*   ngt = (const int*)d_in[4];
  float* out = (float*)d_out;

  const size_t dynBytes =
      (size_t)(NQ * ROWPAD + 2 * NQ) * sizeof(float) + (size_t)(3 * NQ) * sizeof(int);
  const size_t wsNeed = (size_t)NB * NQ * NG * sizeof(float);

  if (d_ws && ws_size >= wsNeed) {
    float* costws = (float*)d_ws;
    const int total = NB * NQ * NG;
    build_cost_kernel<<<(total + 255) / 256, 256, 0, stream>>>(obj, cd, gi, costws);
    hungarian_kernel<true><<<NB, NT, dynBytes, stream>>>(obj, cd, gi, ngt, costws, out);
  } else {
    hungarian_kernel<false><<<NB, NT, dynBytes, stream>>>(obj, cd, gi, ngt, nullptr, out);
  }
}